// ContinuousConvolutionLayer_19155554140235
// MI455X (gfx1250) — compile-verified
//
#include <hip/hip_runtime.h>

// Problem constants (from reference): B=4, S=1024, X=1024, C=8, D=2, WIDTH=16, DEPTH=4
#define B_ 4
#define S_ 1024
#define X_ 1024
#define C_ 8
#define D_ 2
#define DEPTH_ 4
#define SCHUNK 32   // s-values per wave

typedef __attribute__((ext_vector_type(2))) float v2f;
typedef __attribute__((ext_vector_type(8))) float v8f;

// Swap data between lanes l <-> l+16 (wave32).  ds_swizzle group-of-32 mode:
// and_mask=0x1f, or_mask=0, xor_mask=0x10  ->  offset = (0x10<<10) | 0x1f = 0x401F
__device__ __forceinline__ float swap16(float v) {
    return __int_as_float(__builtin_amdgcn_ds_swizzle(__float_as_int(v), 0x401F));
}

// Branch-free tanh: prefer the gfx1250 hardware V_TANH_F32; else a
// divergence-free exp-based formulation (single v_exp + v_rcp).
__device__ __forceinline__ float fast_tanh(float x) {
#if __has_builtin(__builtin_amdgcn_tanhf)
    return __builtin_amdgcn_tanhf(x);
#else
    const float a = __builtin_fabsf(x);
    const float e = __expf(2.0f * a);        // inf-safe: e=inf -> t=1
    const float t = 1.0f - 2.0f / (e + 1.0f);
    return __builtin_copysignf(t, x);
#endif
}

__global__ void cck_zero(float* __restrict__ out, int n) {
    int i = blockIdx.x * blockDim.x + threadIdx.x;
    if (i < n) out[i] = 0.0f;
}

// One wave handles: one (b, x-tile of 16) pair and SCHUNK consecutive s values.
// Activation h kept in WMMA B-layout (lane = point p, feature striped over
// VGPR index + lane half).  Each layer: D = W^T(A) x h^T(B) + bias(C) via
// 4x V_WMMA_F32_16X16X4_F32, then D->B relayout with 8 ds_swizzle + cndmask,
// hardware v_tanh + residual, repeat.  Final einsum over s fused via atomics.
__global__ void __launch_bounds__(256) cck_main(
    const float* __restrict__ yu,   // (B,S,10)
    const float* __restrict__ x,    // (B,X,2)
    const float* __restrict__ W_in, // (1,16)
    const float* __restrict__ b_in, // (16)
    const float* __restrict__ W_hid,// (4,16,16)
    const float* __restrict__ b_hid,// (4,16)
    const float* __restrict__ W_out,// (16,1)
    const float* __restrict__ b_out,// (1)
    float* __restrict__ out)        // (B,X,8)
{
    const int lane  = threadIdx.x & 31;
    const int wave  = threadIdx.x >> 5;
    const int gwave = blockIdx.x * 8 + wave;        // 0 .. 8191
    const int sc    = gwave & 31;                    // s chunk (32 chunks)
    const int xt    = (gwave >> 5) & 63;             // x tile (64 tiles)
    const int b     = gwave >> 11;                   // batch  (4)

    const int p    = lane & 15;                      // point within tile
    const int half = lane >> 4;                      // lane half

    // This lane's x point (lanes l and l+16 share point p).
    const int xi = xt * 16 + p;
    const float xv0 = x[((size_t)b * X_ + xi) * D_ + 0];
    const float xv1 = x[((size_t)b * X_ + xi) * D_ + 1];

    // B-layout feature index for register j: f = 4*(j>>1) + (j&1) + 2*half
    float winl[8], binl[8], woutl[8];
    #pragma unroll
    for (int j = 0; j < 8; ++j) {
        const int f = 4 * (j >> 1) + (j & 1) + 2 * half;
        winl[j]  = W_in[f];
        binl[j]  = b_in[f];
        woutl[j] = W_out[f];
    }

    // Hidden weights in WMMA A-layout as W^T:
    // A element (m=j_out, k=i): VGPR = k%2, lane = m + 16*(k/2)
    // chunk c covers i = 4c..4c+3  ->  wA[L][c].{x,y}[lane] = W_hid[L][4c+{0,1}+2*half][p]
    v2f wA[DEPTH_][4];
    #pragma unroll
    for (int L = 0; L < DEPTH_; ++L)
        #pragma unroll
        for (int c = 0; c < 4; ++c) {
            wA[L][c].x = W_hid[((size_t)L * 16 + (4 * c + 0 + 2 * half)) * 16 + p];
            wA[L][c].y = W_hid[((size_t)L * 16 + (4 * c + 1 + 2 * half)) * 16 + p];
        }

    // Bias in D layout (result is transposed: m = out-feature):
    // VGPR r holds feature r + 8*half
    float bd[DEPTH_][8];
    #pragma unroll
    for (int L = 0; L < DEPTH_; ++L)
        #pragma unroll
        for (int r = 0; r < 8; ++r)
            bd[L][r] = b_hid[L * 16 + r + 8 * half];

    const float bout = b_out[0];
    const float invS = 1.0f / (float)S_;

    // Fused einsum accumulators: this lane owns out[b, xi, 4*half + 0..3]
    float acc0 = 0.f, acc1 = 0.f, acc2 = 0.f, acc3 = 0.f;

    const int s0 = sc * SCHUNK;
    const float* yubase = yu + (size_t)b * S_ * (C_ + D_);

    // Software pipeline (1 deep): issue next s's y/u loads before the MLP.
    float  ny0 = yubase[s0 * (C_ + D_) + C_ + 0];
    float  ny1 = yubase[s0 * (C_ + D_) + C_ + 1];
    float4 nu4 = *(const float4*)(yubase + s0 * (C_ + D_) + 4 * half);

    for (int si = 0; si < SCHUNK; ++si) {
        const float  y0 = ny0, y1 = ny1;
        const float4 u4 = nu4;
        if (si + 1 < SCHUNK) {
            const float* yup = yubase + (s0 + si + 1) * (C_ + D_);
            ny0 = yup[C_ + 0];
            ny1 = yup[C_ + 1];
            nu4 = *(const float4*)(yup + 4 * half);
        }

        const float e0 = xv0 - y0, e1 = xv1 - y1;
        const float r  = e0 * e0 + e1 * e1;

        // h0 in B layout
        float hb[8];
        #pragma unroll
        for (int j = 0; j < 8; ++j) hb[j] = fmaf(r, winl[j], binl[j]);

        #pragma unroll
        for (int L = 0; L < DEPTH_; ++L) {
            // acc (D layout) starts as bias
            v8f acc;
            #pragma unroll
            for (int r2 = 0; r2 < 8; ++r2) acc[r2] = bd[L][r2];

            // pre^T = W^T x h^T : four K=4 chunks of fp32 WMMA
            #pragma unroll
            for (int c = 0; c < 4; ++c) {
                v2f bm; bm.x = hb[2 * c]; bm.y = hb[2 * c + 1];
                acc = __builtin_amdgcn_wmma_f32_16x16x4_f32(
                    false, wA[L][c], false, bm, (short)0, acc, false, false);
            }

            // D layout -> B layout: lane-half swaps + selects
            float d[8], sw[8];
            #pragma unroll
            for (int r2 = 0; r2 < 8; ++r2) { d[r2] = acc[r2]; sw[r2] = swap16(acc[r2]); }
            float nb[8];
            nb[0] = half ? sw[2] : d[0];   // f0 | f2
            nb[1] = half ? sw[3] : d[1];   // f1 | f3
            nb[2] = half ? sw[6] : d[4];   // f4 | f6
            nb[3] = half ? sw[7] : d[5];   // f5 | f7
            nb[4] = half ? d[2]  : sw[0];  // f8 | f10
            nb[5] = half ? d[3]  : sw[1];  // f9 | f11
            nb[6] = half ? d[6]  : sw[4];  // f12| f14
            nb[7] = half ? d[7]  : sw[5];  // f13| f15

            // residual: h = tanh(pre) + h   (elementwise, both in B layout)
            #pragma unroll
            for (int j = 0; j < 8; ++j) hb[j] = fast_tanh(nb[j]) + hb[j];
        }

        // k[p] = h[p] . W_out + b_out ; each lane holds 8 of 16 features,
        // partner lane (l^16) holds the other 8 -> swap & add.
        float kp = 0.f;
        #pragma unroll
        for (int j = 0; j < 8; ++j) kp = fmaf(hb[j], woutl[j], kp);
        const float kfull = (kp + swap16(kp) + bout) * invS;

        // fused einsum: out[b,xi,c] += k * u[b,s,c]
        acc0 = fmaf(kfull, u4.x, acc0);
        acc1 = fmaf(kfull, u4.y, acc1);
        acc2 = fmaf(kfull, u4.z, acc2);
        acc3 = fmaf(kfull, u4.w, acc3);
    }

    float* op = out + ((size_t)b * X_ + xi) * C_ + 4 * half;
    atomicAdd(op + 0, acc0);
    atomicAdd(op + 1, acc1);
    atomicAdd(op + 2, acc2);
    atomicAdd(op + 3, acc3);
}

extern "C" void kernel_launch(void* const* d_in, const int* in_sizes, int n_in,
                              void* d_out, int out_size, void* d_ws, size_t ws_size,
                              hipStream_t stream) {
    (void)in_sizes; (void)n_in; (void)d_ws; (void)ws_size;
    const float* yu    = (const float*)d_in[0];
    const float* x     = (const float*)d_in[1];
    const float* W_in  = (const float*)d_in[2];
    const float* b_in  = (const float*)d_in[3];
    const float* W_hid = (const float*)d_in[4];
    const float* b_hid = (const float*)d_in[5];
    const float* W_out = (const float*)d_in[6];
    const float* b_out = (const float*)d_in[7];
    float* out = (float*)d_out;

    const int n_out = B_ * X_ * C_;  // 32768
    cck_zero<<<(n_out + 255) / 256, 256, 0, stream>>>(out, n_out);

    // 4 b * 64 x-tiles * 32 s-chunks = 8192 waves / 8 waves-per-block
    cck_main<<<1024, 256, 0, stream>>>(yu, x, W_in, b_in, W_hid, b_hid,
                                       W_out, b_out, out);
}